// JSDistanceLoss_17102559773153
// MI455X (gfx1250) — compile-verified
//
#include <hip/hip_runtime.h>
#include <stdint.h>

#define ALPHA_F 0.5f
#define LAM_F 0.9f
#define TEMP_F 1.0f
#define IGNORE_INDEX_I (-100)
#define THREADS 256

// ---------------- wave32 / block reductions (deterministic) ----------------

__device__ __forceinline__ float waveSum(float v) {
#pragma unroll
  for (int o = 16; o > 0; o >>= 1) v += __shfl_xor(v, o, 32);
  return v;
}

__device__ __forceinline__ float waveMax(float v) {
#pragma unroll
  for (int o = 16; o > 0; o >>= 1) v = fmaxf(v, __shfl_xor(v, o, 32));
  return v;
}

// All threads return the same value; fixed accumulation order -> deterministic.
__device__ __forceinline__ float blockSum(float v, float* scr) {
  const int lane = threadIdx.x & 31;
  const int w = threadIdx.x >> 5;
  const int nw = (int)(blockDim.x >> 5);
  v = waveSum(v);
  __syncthreads();                 // protect scr from previous use
  if (lane == 0) scr[w] = v;
  __syncthreads();
  float r = scr[0];
  for (int i = 1; i < nw; ++i) r += scr[i];
  return r;
}

__device__ __forceinline__ float blockMax(float v, float* scr) {
  const int lane = threadIdx.x & 31;
  const int w = threadIdx.x >> 5;
  const int nw = (int)(blockDim.x >> 5);
  v = waveMax(v);
  __syncthreads();
  if (lane == 0) scr[w] = v;
  __syncthreads();
  float r = scr[0];
  for (int i = 1; i < nw; ++i) r = fmaxf(r, scr[i]);
  return r;
}

// ---------------- kernel 1: one workgroup per token row ----------------
// LDS layout: [V floats student][V floats teacher]  (V=32000 -> 256000 B <= 320 KB WGP LDS)
// Row is DMA'd from HBM exactly once via GLOBAL_LOAD_ASYNC_TO_LDS_B128 (ASYNCcnt path),
// with a non-temporal hint since every byte is single-use (1.05 GB stream vs 192 MB L2).

__global__ void __launch_bounds__(THREADS)
jsd_row_kernel(const float* __restrict__ s_logits,
               const float* __restrict__ t_logits,
               const int* __restrict__ labels,
               float* __restrict__ ws,
               int rows, int V)
{
  extern __shared__ char smem[];
  __shared__ float scr[THREADS / 32];

  const int row = blockIdx.x;
  if (row >= rows) return;

  const float* srow = s_logits + (size_t)row * (size_t)V;
  const float* trow = t_logits + (size_t)row * (size_t)V;

  float* sl = (float*)smem;       // student row in LDS
  float* tl = ((float*)smem) + V; // teacher row in LDS

  // Hardware truncates a generic LDS address to addr[31:0] as the LDS byte
  // offset (ISA 10.2 aperture rules), so the low 32 bits of the generic
  // pointer are exactly what the async-load VDST operand wants.
  const uint32_t ldsBase = (uint32_t)(uintptr_t)smem;
  const uint32_t VB = (uint32_t)V * 4u;   // bytes per row (V % 4 == 0 here)

  // ---- async stage: global -> LDS, 16B per lane per op (GVS addressing) ----
  for (uint32_t c = (uint32_t)threadIdx.x * 16u; c < VB; c += THREADS * 16u) {
    asm volatile("global_load_async_to_lds_b128 %0, %1, %2 th:TH_LOAD_NT"
                 :: "v"(ldsBase + c), "v"(c), "s"(srow) : "memory");
    asm volatile("global_load_async_to_lds_b128 %0, %1, %2 th:TH_LOAD_NT"
                 :: "v"(ldsBase + VB + c), "v"(c), "s"(trow) : "memory");
  }
  asm volatile("s_wait_asynccnt 0" ::: "memory");
  __syncthreads();

  const float4* s4 = (const float4*)sl;
  const float4* t4 = (const float4*)tl;
  const int V4 = V >> 2;
  const int Vtail = V4 << 2;

  // ---- pass A: row maxima (T==1 so scaled logits == raw logits) ----
  float ms = -__builtin_inff(), mt = -__builtin_inff();
  for (int i = threadIdx.x; i < V4; i += THREADS) {
    const float4 a = s4[i];
    const float4 b = t4[i];
    ms = fmaxf(ms, fmaxf(fmaxf(a.x, a.y), fmaxf(a.z, a.w)));
    mt = fmaxf(mt, fmaxf(fmaxf(b.x, b.y), fmaxf(b.z, b.w)));
  }
  for (int i = Vtail + threadIdx.x; i < V; i += THREADS) {
    ms = fmaxf(ms, sl[i]);
    mt = fmaxf(mt, tl[i]);
  }
  ms = blockMax(ms, scr);
  mt = blockMax(mt, scr);

  // ---- pass B: partition functions ----
  float zs = 0.0f, zt = 0.0f;
  for (int i = threadIdx.x; i < V4; i += THREADS) {
    const float4 a = s4[i];
    const float4 b = t4[i];
    zs += __expf(a.x - ms) + __expf(a.y - ms) + __expf(a.z - ms) + __expf(a.w - ms);
    zt += __expf(b.x - mt) + __expf(b.y - mt) + __expf(b.z - mt) + __expf(b.w - mt);
  }
  for (int i = Vtail + threadIdx.x; i < V; i += THREADS) {
    zs += __expf(sl[i] - ms);
    zt += __expf(tl[i] - mt);
  }
  zs = blockSum(zs, scr);
  zt = blockSum(zt, scr);
  const float lnZs = __logf(zs), lnZt = __logf(zt);
  const float izs = 1.0f / zs, izt = 1.0f / zt;

  // ---- pass C: sum_v p_s*(log m - log p_s), p_t*(log m - log p_t) ----
  float xs = 0.0f, xt = 0.0f;
#define JSD_TERM(av, bv)                                                   \
  {                                                                        \
    const float ps_ = __expf((av)-ms) * izs;                               \
    const float pt_ = __expf((bv)-mt) * izt;                               \
    const float lm_ = __logf(LAM_F * ps_ + (1.0f - LAM_F) * pt_);          \
    if (!(__builtin_isinf(av) || __builtin_isinf(bv))) {                   \
      xs += ps_ * (lm_ - (((av)-ms) - lnZs));                              \
      xt += pt_ * (lm_ - (((bv)-mt) - lnZt));                              \
    }                                                                      \
  }
  for (int i = threadIdx.x; i < V4; i += THREADS) {
    const float4 a = s4[i];
    const float4 b = t4[i];
    JSD_TERM(a.x, b.x)
    JSD_TERM(a.y, b.y)
    JSD_TERM(a.z, b.z)
    JSD_TERM(a.w, b.w)
  }
  for (int i = Vtail + threadIdx.x; i < V; i += THREADS) {
    JSD_TERM(sl[i], tl[i])
  }
#undef JSD_TERM
  xs = blockSum(xs, scr);
  xt = blockSum(xt, scr);

  if (threadIdx.x == 0) {
    const int lab = labels[row];
    const float mk = (lab != IGNORE_INDEX_I) ? 1.0f : 0.0f;
    const int safe = (lab == IGNORE_INDEX_I) ? 0 : lab;
    const float picked = (sl[safe] - ms) - lnZs;  // log_softmax(student)[label]
    ws[row]            = xs * mk;
    ws[rows + row]     = xt * mk;
    ws[2 * rows + row] = picked * mk;
    ws[3 * rows + row] = mk;
  }
}

// ---------------- kernel 2: deterministic scalar combine ----------------

__global__ void __launch_bounds__(THREADS)
jsd_finalize_kernel(const float* __restrict__ ws, float* __restrict__ out, int rows)
{
  __shared__ float scr[THREADS / 32];
  float sxs = 0.0f, sxt = 0.0f, spk = 0.0f, smk = 0.0f;
  for (int i = threadIdx.x; i < rows; i += THREADS) {
    sxs += ws[i];
    sxt += ws[rows + i];
    spk += ws[2 * rows + i];
    smk += ws[3 * rows + i];
  }
  sxs = blockSum(sxs, scr);
  sxt = blockSum(sxt, scr);
  spk = blockSum(spk, scr);
  smk = blockSum(smk, scr);
  if (threadIdx.x == 0) {
    const float n = smk;
    float distil = LAM_F * (-sxs / n) + (1.0f - LAM_F) * (-sxt / n);
    distil *= TEMP_F * TEMP_F;
    const float hard = -spk / n;
    out[0] = ALPHA_F * distil + (1.0f - ALPHA_F) * hard;
  }
}

// ---------------- host entry ----------------

extern "C" void kernel_launch(void* const* d_in, const int* in_sizes, int n_in,
                              void* d_out, int out_size, void* d_ws, size_t ws_size,
                              hipStream_t stream) {
  const float* s = (const float*)d_in[0];
  const float* t = (const float*)d_in[1];
  const int* labels = (const int*)d_in[2];
  float* out = (float*)d_out;
  float* ws = (float*)d_ws;

  const int rows = in_sizes[2];            // B*S = 4096
  const int V = in_sizes[0] / rows;        // 32000

  const size_t shmem = (size_t)2 * (size_t)V * sizeof(float);  // 256000 B

  jsd_row_kernel<<<rows, THREADS, shmem, stream>>>(s, t, labels, ws, rows, V);
  jsd_finalize_kernel<<<1, THREADS, 0, stream>>>(ws, out, rows);
}